// InteractLayer_30760555774312
// MI455X (gfx1250) — compile-verified
//
#include <hip/hip_runtime.h>
#include <math.h>

// Reference: InteractLayer — B=2048, S=201, C=768, N_USERS=100000, fp32.
// Memory-bound (~3.2 GB total traffic -> ~140us @ 23.3 TB/s). Compute
// (two 2048x768x768 fp32 GEMMs used only as quadratic forms) is negligible,
// done with V_WMMA_F32_16X16X4_F32 to keep full fp32 precision.

typedef __attribute__((ext_vector_type(2))) float v2f;
typedef __attribute__((ext_vector_type(8))) float v8f;

#define B_BATCH 2048
#define SEQ     201
#define CH      768
#define SC      (SEQ * CH)      // 154368
#define TILE_M  16
#define NTILES  (CH / 16)       // 48 N-tiles of the 768-wide output
#define N_USERS 100000

// One block = 16 batch rows, 256 threads = 8 waves (wave32).
// LDS: dynamic 2*16*768 floats (96 KB) for T (text_ini) and G (graph_ini).
__global__ __launch_bounds__(256)
void interact_fused_kernel(const float* __restrict__ text,
                           const float* __restrict__ auf,
                           const int*   __restrict__ user_index,
                           const float* __restrict__ Wt,
                           const float* __restrict__ bt,
                           const float* __restrict__ Wg,
                           const float* __restrict__ bg,
                           float* __restrict__ out_text,
                           float* __restrict__ out_user)
{
    __shared__ float redA[TILE_M];   // a = t·(Wt t) + t·bt
    __shared__ float redB[TILE_M];   // b = g·t  (also 'd')
    __shared__ float redC[TILE_M];   // c = g·(Wg g) + g·bg
    extern __shared__ float smem[];
    float* Tl = smem;                 // 16 x 768
    float* Gl = smem + TILE_M * CH;   // 16 x 768

    const int tid  = threadIdx.x;
    const int lane = tid & 31;
    const int wave = tid >> 5;
    const int base = blockIdx.x * TILE_M;

    if (tid < TILE_M) { redA[tid] = 0.f; redB[tid] = 0.f; redC[tid] = 0.f; }
    __syncthreads();

    // ---- Phase 0: gather rows into LDS; bias-dots and g·t dot via ds_add_f32.
    for (int m = 0; m < TILE_M; ++m) {
        const float* trow = text + (size_t)(base + m) * SC;            // s = 0
        const float* grow = auf  + (size_t)user_index[base + m] * CH;
        float pa = 0.f, pb = 0.f, pc = 0.f;
        #pragma unroll
        for (int j = 0; j < 3; ++j) {      // 768 = 3 * 256
            int c = tid + j * 256;
            float t = trow[c];
            float g = grow[c];
            Tl[m * CH + c] = t;
            Gl[m * CH + c] = g;
            pa += t * bt[c];
            pb += g * t;
            pc += g * bg[c];
        }
        atomicAdd(&redA[m], pa);
        atomicAdd(&redB[m], pb);
        atomicAdd(&redC[m], pc);
    }
    __syncthreads();

    // A 16x4 f32 layout: lanes 0-15 -> M=lane, VGPR0/1 = K0/K1;
    //                    lanes 16-31 -> M=lane-16, VGPR0/1 = K2/K3.
    // B 4x16 mirrors it with N across lanes. C/D: lane l, vgpr v ->
    //                    M = v + 8*(l>=16), N = l%16.
    const int arow = lane & 15;
    const int koff = (lane >> 4) * 2;
    const int mhi  = (lane >> 4) << 3;

    // ---- Phase 1: text_tmp = T @ Wt^T via f32 WMMA, fused dot into redA.
    for (int nt = wave; nt < NTILES; nt += 8) {
        v8f acc = {};
        #pragma unroll 8
        for (int k0 = 0; k0 < CH; k0 += 4) {
            v2f a = *(const v2f*)&Tl[arow * CH + k0 + koff];
            v2f b = *(const v2f*)&Wt[(size_t)(nt * 16 + arow) * CH + k0 + koff];
            acc = __builtin_amdgcn_wmma_f32_16x16x4_f32(
                false, a, false, b, (short)0, acc, false, false);
        }
        int ncol = nt * 16 + arow;
        #pragma unroll
        for (int v = 0; v < 8; ++v) {
            int m = v + mhi;
            atomicAdd(&redA[m], acc[v] * Tl[m * CH + ncol]);
        }
    }

    // ---- Phase 2: graph_tmp = G @ Wg^T via f32 WMMA, fused dot into redC.
    for (int nt = wave; nt < NTILES; nt += 8) {
        v8f acc = {};
        #pragma unroll 8
        for (int k0 = 0; k0 < CH; k0 += 4) {
            v2f a = *(const v2f*)&Gl[arow * CH + k0 + koff];
            v2f b = *(const v2f*)&Wg[(size_t)(nt * 16 + arow) * CH + k0 + koff];
            acc = __builtin_amdgcn_wmma_f32_16x16x4_f32(
                false, a, false, b, (short)0, acc, false, false);
        }
        int ncol = nt * 16 + arow;
        #pragma unroll
        for (int v = 0; v < 8; ++v) {
            int m = v + mhi;
            atomicAdd(&redC[m], acc[v] * Gl[m * CH + ncol]);
        }
    }
    __syncthreads();

    // ---- Phase 3: 2-way softmaxes, blend, scatter to outputs.
    for (int e = tid; e < TILE_M * CH; e += 256) {
        int m = e / CH, c = e % CH;
        float t = Tl[e], g = Gl[e];

        float a = redA[m], b = redB[m];
        float mx = fmaxf(a, b);
        float ea = __expf(a - mx), eb = __expf(b - mx);
        float inv = 1.0f / (ea + eb);
        out_text[(size_t)(base + m) * SC + c] = (ea * inv) * t + (eb * inv) * g;

        float cc = redC[m], d = redB[m];
        float mx2 = fmaxf(cc, d);
        float ec = __expf(cc - mx2), ed = __expf(d - mx2);
        float inv2 = 1.0f / (ec + ed);
        out_user[(size_t)user_index[base + m] * CH + c] =
            (ec * inv2) * g + (ed * inv2) * t;
    }
}

extern "C" void kernel_launch(void* const* d_in, const int* in_sizes, int n_in,
                              void* d_out, int out_size, void* d_ws, size_t ws_size,
                              hipStream_t stream) {
    const float* text = (const float*)d_in[0];
    const float* auf  = (const float*)d_in[1];
    const int*   uidx = (const int*)d_in[2];
    const float* Wt   = (const float*)d_in[3];
    const float* bt   = (const float*)d_in[4];
    const float* Wg   = (const float*)d_in[5];
    const float* bg   = (const float*)d_in[6];

    float* out_text = (float*)d_out;
    float* out_user = out_text + (size_t)B_BATCH * SC;

    // Bulk pass-through data: stream-ordered D2D copies saturate HBM.
    hipMemcpyAsync(out_text, text, (size_t)B_BATCH * SC * sizeof(float),
                   hipMemcpyDeviceToDevice, stream);
    hipMemcpyAsync(out_user, auf, (size_t)N_USERS * CH * sizeof(float),
                   hipMemcpyDeviceToDevice, stream);

    // Fused gather + WMMA quadratic forms + softmax blend + scatter;
    // overwrites s=0 text rows and the 2048 gathered user rows (stream order).
    size_t shmem = (size_t)2 * TILE_M * CH * sizeof(float);
    interact_fused_kernel<<<B_BATCH / TILE_M, 256, shmem, stream>>>(
        text, auf, uidx, Wt, bt, Wg, bg, out_text, out_user);
}